// CollectiveCenCriticPoliceFull_33328946217261
// MI455X (gfx1250) — compile-verified
//
#include <hip/hip_runtime.h>

typedef __attribute__((ext_vector_type(2))) float v2f;
typedef __attribute__((ext_vector_type(4))) float v4f;
typedef __attribute__((ext_vector_type(8))) float v8f;

#define B_ 4096
#define Z_ 128
#define C_ 12
#define H_ 512

// ---------------------------------------------------------------------------
// gfx1250 async global->LDS copy (ASYNCcnt-tracked, no staging VGPRs).
// GVS addressing: 64-bit SGPR base + 32-bit per-lane byte offset.
// ---------------------------------------------------------------------------
__device__ __forceinline__ void async_b32(unsigned lds_byte, unsigned gl_byte,
                                          const float* base) {
  asm volatile("global_load_async_to_lds_b32 %0, %1, %2"
               :: "v"(lds_byte), "v"(gl_byte), "s"(base) : "memory");
}
__device__ __forceinline__ void wait_async0() {
  asm volatile("s_wait_asynccnt 0x0" ::: "memory");
}
__device__ __forceinline__ unsigned lds_off(const float* p) {
  // generic LDS pointer: addr[31:0] is the LDS byte offset the HW uses
  return (unsigned)(size_t)p;
}

// ---------------------------------------------------------------------------
// Kernel 1: edge weights M[j,i] = adj[j,i] * exp(-w0 * tt[j,i])   (128x128)
// ---------------------------------------------------------------------------
__global__ __launch_bounds__(256) void edge_kernel(const int* __restrict__ adj,
                                                   const int* __restrict__ tt,
                                                   const float* __restrict__ w0,
                                                   float* __restrict__ Mw) {
  int idx = blockIdx.x * 256 + threadIdx.x;  // 16384 total
  float w0v = w0[0];
  Mw[idx] = adj[idx] ? __expf(-w0v * (float)tt[idx]) : 0.0f;
}

// ---------------------------------------------------------------------------
// Kernel 2 (the roofline: ~293MB stream at 23.3 TB/s).
// Block = 128 threads = 4 waves; wave owns one b, lane owns 4 consecutive i
// -> one float4 per lane per j-row (512B fully-coalesced rows). Read-once
// streams (action_count/obs) use nontemporal loads so they don't evict the
// L2-resident GEMM weights.
// ---------------------------------------------------------------------------
__global__ __launch_bounds__(128) void next_kernel(const float* __restrict__ obs,
                                                   const float* __restrict__ ac,
                                                   const float* __restrict__ Mw,
                                                   const float* __restrict__ w0,
                                                   float* __restrict__ nextOut) {
  const int sub  = threadIdx.x >> 5;
  const int lane = threadIdx.x & 31;
  const int b    = blockIdx.x * 4 + sub;
  const int i0   = lane * 4;

  const float r = __expf(-w0[0]);
  float w[C_];
  w[0] = 1.0f;
#pragma unroll
  for (int c = 1; c < C_; ++c) w[c] = w[c - 1] * r;

  const v4f* ob4 = (const v4f*)(obs + ((size_t)b * Z_ + i0) * C_);
  v4f s = {};
#pragma unroll
  for (int ii = 0; ii < 4; ++ii) {
    float t = 0.0f;
#pragma unroll
    for (int q = 0; q < 3; ++q) {
      v4f o = __builtin_nontemporal_load(ob4 + ii * 3 + q);
      t += o[0] * w[q * 4 + 0] + o[1] * w[q * 4 + 1] +
           o[2] * w[q * 4 + 2] + o[3] * w[q * 4 + 3];
    }
    s[ii] = t;
  }

  const float* acb = ac + (size_t)b * Z_ * Z_ + i0;
  const float* mwp = Mw + i0;
#pragma unroll 4
  for (int j = 0; j < Z_; ++j) {
    v4f a = __builtin_nontemporal_load((const v4f*)(acb + (size_t)j * Z_));
    v4f m = *(const v4f*)(mwp + j * Z_);
    s += a * m;
  }
  *(v4f*)(nextOut + (size_t)b * Z_ + i0) = s;
}

// ---------------------------------------------------------------------------
// Kernel 3: fp32 WMMA GEMM  D = relu?(A[MxK] * W[KxN] + bias)
// 8 waves/block, 128x64 block tile, wave = 16 rows x four 16x16 acc tiles.
// V_WMMA_F32_16X16X4_F32 (full fp32).
//  - A tile: rows padded to 17 floats -> conflict-free a-pair ds_load_2addr.
//  - B tile: K-pair interleaved Bs[p][n][2] (p=k/2, row stride 160 dwords):
//    each B operand is ONE aligned ds_load_b64 into an even VGPR pair (no
//    repacking movs); lower/upper lane halves hit banks 0-31 / 32-63.
//  - Double-buffered GLOBAL_LOAD_ASYNC_TO_LDS_B32 staging: tile t+1 streams
//    in while the 16 WMMAs of tile t execute; s_wait_asynccnt 0 gates only
//    the tile being consumed.
// ---------------------------------------------------------------------------
template <int K, int N, bool RELU>
__global__ __launch_bounds__(256) void gemm_wmma(const float* __restrict__ A,
                                                 const float* __restrict__ Bm,
                                                 const float* __restrict__ bias,
                                                 float* __restrict__ D) {
  __shared__ float As[2][128 * 17];  // 128 rows x 16 (+1 pad)
  __shared__ float Bs[2][8 * 160];   // 8 k-pairs x 64 n x 2 (+pad to 160)

  const int tid  = threadIdx.x;
  const int wave = tid >> 5;
  const int lane = tid & 31;
  const int m16  = lane & 15;
  const int hi   = lane >> 4;         // 0 / 1
  const int hi2  = hi << 1;           // 0 / 2
  const int row0 = blockIdx.x * 128;
  const int col0 = blockIdx.y * 64;

  // staging coordinates (constant across k tiles)
  const int ar0 = (tid >> 2);
  const int ac0 = (tid & 3) << 2;
  const int ar1 = ((tid + 256) >> 2);
  const int br  = tid >> 4;           // B row (k within tile), 0..15
  const int bc  = (tid & 15) << 2;    // B col4

  unsigned ldsA0[2], ldsA1[2], ldsB[2];
#pragma unroll
  for (int s = 0; s < 2; ++s) {
    ldsA0[s] = lds_off(&As[s][ar0 * 17 + ac0]);
    ldsA1[s] = lds_off(&As[s][ar1 * 17 + ac0]);
    ldsB[s]  = lds_off(&Bs[s][(br >> 1) * 160 + bc * 2 + (br & 1)]);
  }

  auto stage = [&](int k0, int sel) {
    unsigned ga0 = (unsigned)(((row0 + ar0) * K + k0 + ac0) * 4);
    unsigned ga1 = (unsigned)(((row0 + ar1) * K + k0 + ac0) * 4);
#pragma unroll
    for (int e = 0; e < 4; ++e) {
      async_b32(ldsA0[sel] + 4u * e, ga0 + 4u * e, A);
      async_b32(ldsA1[sel] + 4u * e, ga1 + 4u * e, A);
    }
    unsigned gb = (unsigned)(((k0 + br) * N + col0 + bc) * 4);
#pragma unroll
    for (int e = 0; e < 4; ++e) {  // dest stride: 2 dwords per successive col
      async_b32(ldsB[sel] + 8u * e, gb + 4u * e, Bm);
    }
  };

  v8f acc[4] = {v8f{}, v8f{}, v8f{}, v8f{}};

  constexpr int NT = K / 16;
  stage(0, 0);
  for (int t = 0; t < NT; ++t) {
    const int sel = t & 1;
    wait_async0();
    __syncthreads();
    if (t + 1 < NT) stage((t + 1) * 16, sel ^ 1);

    const float* Asb = &As[sel][0];
    const float* Bsb = &Bs[sel][0];
    const int aRow = (wave * 16 + m16) * 17;
#pragma unroll
    for (int ks = 0; ks < 4; ++ks) {
      const int k = ks * 4 + hi2;     // even; K-pair index p = k/2
      const int p = ks * 2 + hi;
      // A operand: VGPR0 = K{0|2}, VGPR1 = K{1|3}, lane%16 = M
      v2f a;
      a.x = Asb[aRow + k];
      a.y = Asb[aRow + k + 1];
#pragma unroll
      for (int nt = 0; nt < 4; ++nt) {
        // B operand: single aligned b64 load of {B[k][n], B[k+1][n]}
        v2f b = *(const v2f*)(Bsb + p * 160 + (nt * 16 + m16) * 2);
        acc[nt] = __builtin_amdgcn_wmma_f32_16x16x4_f32(
            false, a, false, b, (short)0, acc[nt], false, false);
      }
    }
    __syncthreads();
  }

  // epilogue: VGPR r, lane l -> M = r + 8*(l>=16), N = l%16
  float bia[4];
#pragma unroll
  for (int nt = 0; nt < 4; ++nt) bia[nt] = bias[col0 + nt * 16 + m16];

  const int mBase = row0 + wave * 16 + hi * 8;
#pragma unroll
  for (int nt = 0; nt < 4; ++nt) {
#pragma unroll
    for (int r = 0; r < 8; ++r) {
      float v = acc[nt][r] + bia[nt];
      if (RELU) v = v > 0.0f ? v : 0.0f;
      D[(size_t)(mBase + r) * N + col0 + nt * 16 + m16] = v;
    }
  }
}

// ---------------------------------------------------------------------------
// Kernel 4: val[b] = x2[b,:] . Wf + bf   (wave per row, float4 + shfl)
// ---------------------------------------------------------------------------
__global__ __launch_bounds__(256) void head_kernel(const float* __restrict__ X2,
                                                   const float* __restrict__ Wf,
                                                   const float* __restrict__ bf,
                                                   float* __restrict__ val) {
  const int row  = blockIdx.x * 8 + (threadIdx.x >> 5);
  const int lane = threadIdx.x & 31;
  const v4f* xr = (const v4f*)(X2 + (size_t)row * H_);
  const v4f* wf = (const v4f*)Wf;
  float s = 0.0f;
#pragma unroll
  for (int t = 0; t < H_ / 128; ++t) {
    v4f x = xr[lane + 32 * t];
    v4f w = wf[lane + 32 * t];
    s += x[0] * w[0] + x[1] * w[1] + x[2] * w[2] + x[3] * w[3];
  }
#pragma unroll
  for (int off = 16; off >= 1; off >>= 1) s += __shfl_xor(s, off, 32);
  if (lane == 0) val[row] = s + bf[0];
}

// ---------------------------------------------------------------------------
extern "C" void kernel_launch(void* const* d_in, const int* in_sizes, int n_in,
                              void* d_out, int out_size, void* d_ws, size_t ws_size,
                              hipStream_t stream) {
  (void)in_sizes; (void)n_in; (void)out_size; (void)ws_size;
  const float* obs = (const float*)d_in[0];
  const float* ac  = (const float*)d_in[1];
  const float* w0  = (const float*)d_in[2];
  const float* W1  = (const float*)d_in[3];
  const float* b1  = (const float*)d_in[4];
  const float* W2  = (const float*)d_in[5];
  const float* b2  = (const float*)d_in[6];
  const float* Wf  = (const float*)d_in[7];
  const float* bf  = (const float*)d_in[8];
  const int*   adj = (const int*)d_in[9];
  const int*   tt  = (const int*)d_in[10];

  float* out   = (float*)d_out;
  float* val   = out;        // [4096]
  float* nextS = out + B_;   // [4096*128]

  float* ws = (float*)d_ws;
  float* Mw = ws;                         // 16384 floats
  float* X1 = ws + 16384;                 // 4096*512
  float* X2 = X1 + (size_t)B_ * H_;       // 4096*512

  edge_kernel<<<64, 256, 0, stream>>>(adj, tt, w0, Mw);
  next_kernel<<<B_ / 4, 128, 0, stream>>>(obs, ac, Mw, w0, nextS);
  gemm_wmma<Z_, H_, true><<<dim3(B_ / 128, H_ / 64), 256, 0, stream>>>(nextS, W1, b1, X1);
  gemm_wmma<H_, H_, true><<<dim3(B_ / 128, H_ / 64), 256, 0, stream>>>(X1, W2, b2, X2);
  head_kernel<<<B_ / 8, 256, 0, stream>>>(X2, Wf, bf, val);
}